// EncoderCond_79869211836484
// MI455X (gfx1250) — compile-verified
//
#include <hip/hip_runtime.h>

// ---------------------------------------------------------------------------
// Conditional-VGAE GCN encoder for gfx1250 (MI455X).
// Dense x@W done with V_WMMA_F32_16X16X4_F32 (fp32 WMMA, 16x16 tiles).
// Edge aggregation done edge-parallel with float4 gathers + f32 atomic adds
// (working set ~80MB -> resident in the 192MB L2, so atomics hit L2).
// ---------------------------------------------------------------------------

typedef __attribute__((ext_vector_type(2))) float v2f;
typedef __attribute__((ext_vector_type(8))) float v8f;

__device__ __forceinline__ v8f wmma_f32x4(v2f a, v2f b, v8f c) {
  // 8 args: (neg_a, A, neg_b, B, c_mod, C, reuse_a, reuse_b)
  return __builtin_amdgcn_wmma_f32_16x16x4_f32(false, a, false, b, (short)0, c,
                                               false, false);
}

// ---------------------------- small utilities ------------------------------

__global__ void fill_kernel(float* __restrict__ p, float v, int n) {
  int t = blockIdx.x * blockDim.x + threadIdx.x;
  if (t < n) p[t] = v;
}

__global__ void deg_scatter_kernel(const int* __restrict__ dst,
                                   float* __restrict__ deg, int E) {
  int t = blockIdx.x * blockDim.x + threadIdx.x;
  if (t < E) atomicAdd(&deg[dst[t]], 1.0f);
}

__global__ void rsqrt_kernel(float* __restrict__ p, int n) {
  int t = blockIdx.x * blockDim.x + threadIdx.x;
  if (t < n) p[t] = rsqrtf(p[t]);  // deg >= 1 always (self loops)
}

__global__ void relu_kernel(float* __restrict__ p, int n) {
  int t = blockIdx.x * blockDim.x + threadIdx.x;
  if (t < n) p[t] = fmaxf(p[t], 0.0f);
}

// ------------------------------- GEMMs -------------------------------------
// One wave computes a 16-row x (NT*16)-col tile of C = A x W.
// A layout (16x4 f32): lane<16: {K0,K1}; lane>=16: {K2,K3}  (same M = lane&15)
// B layout (4x16 f32): mirrored with N = lane&15
// C/D: 8 VGPRs, element r -> (M = r + 8*(lane>=16), N = lane&15)

// Layer 1: A = [x | c] (K = 12), W1 [12 x 64], out [N x 64]
__global__ void gemm1_kernel(const float* __restrict__ x,
                             const float* __restrict__ c,
                             const float* __restrict__ W,
                             float* __restrict__ out, int ntiles) {
  int gw = (blockIdx.x * blockDim.x + threadIdx.x) >> 5;
  int lane = threadIdx.x & 31;
  if (gw >= ntiles) return;  // wave-uniform exit, EXEC stays all-1 in WMMA
  int half = lane >> 4;
  int lm = lane & 15;
  int m = gw * 16 + lm;
  const float* xr = x + (size_t)m * 8;

  v2f a[3];
  a[0] = {xr[2 * half + 0], xr[2 * half + 1]};
  a[1] = {xr[2 * half + 4], xr[2 * half + 5]};
  a[2] = {c[2 * half + 0], c[2 * half + 1]};  // broadcast condition vec

  v8f acc[4] = {};
#pragma unroll
  for (int kc = 0; kc < 3; ++kc) {
    int k0 = 4 * kc + 2 * half;
#pragma unroll
    for (int j = 0; j < 4; ++j) {
      v2f b = {W[(size_t)k0 * 64 + j * 16 + lm],
               W[(size_t)(k0 + 1) * 64 + j * 16 + lm]};
      acc[j] = wmma_f32x4(a[kc], b, acc[j]);
    }
  }
#pragma unroll
  for (int j = 0; j < 4; ++j)
#pragma unroll
    for (int r = 0; r < 8; ++r)
      out[(size_t)(gw * 16 + r + 8 * half) * 64 + j * 16 + lm] = acc[j][r];
}

// Generic K=64 layer: A [N x 64], W [64 x NT*16], out [N x NT*16]
template <int NT>
__global__ void gemm_k64_kernel(const float* __restrict__ A,
                                const float* __restrict__ W,
                                float* __restrict__ out, int ntiles) {
  int gw = (blockIdx.x * blockDim.x + threadIdx.x) >> 5;
  int lane = threadIdx.x & 31;
  if (gw >= ntiles) return;
  const int LD = NT * 16;
  int half = lane >> 4;
  int lm = lane & 15;
  const float* ar = A + (size_t)(gw * 16 + lm) * 64;

  v8f acc[NT] = {};
#pragma unroll
  for (int kc = 0; kc < 16; ++kc) {
    int k0 = 4 * kc + 2 * half;
    v2f a = {ar[k0], ar[k0 + 1]};
#pragma unroll
    for (int j = 0; j < NT; ++j) {
      v2f b = {W[(size_t)k0 * LD + j * 16 + lm],
               W[(size_t)(k0 + 1) * LD + j * 16 + lm]};
      acc[j] = wmma_f32x4(a, b, acc[j]);
    }
  }
#pragma unroll
  for (int j = 0; j < NT; ++j)
#pragma unroll
    for (int r = 0; r < 8; ++r)
      out[(size_t)(gw * 16 + r + 8 * half) * LD + j * 16 + lm] = acc[j][r];
}

// --------------------------- GCN aggregation -------------------------------
// out[i] = bias + dinv[i]^2 * h[i]   (self-loop + bias folded in)
template <int D>
__global__ void agg_init_kernel(const float* __restrict__ h,
                                const float* __restrict__ dinv,
                                const float* __restrict__ bias,
                                float* __restrict__ out, int N) {
  const int Q = D / 4;
  int t = blockIdx.x * blockDim.x + threadIdx.x;
  int i = t / Q, q = t % Q;
  if (i >= N) return;
  float s = dinv[i];
  float w = s * s;
  float4 hv = ((const float4*)(h + (size_t)i * D))[q];
  float4 bv = ((const float4*)bias)[q];
  float4 o = {bv.x + w * hv.x, bv.y + w * hv.y, bv.z + w * hv.z,
              bv.w + w * hv.w};
  ((float4*)(out + (size_t)i * D))[q] = o;
}

// out[dst] += dinv[src]*dinv[dst] * h[src]  over all real edges
template <int D>
__global__ void scatter_kernel(const int* __restrict__ src,
                               const int* __restrict__ dst,
                               const float* __restrict__ dinv,
                               const float* __restrict__ h,
                               float* __restrict__ out, int E) {
  const int Q = D / 4;
  long long t = (long long)blockIdx.x * blockDim.x + threadIdx.x;
  int e = (int)(t / Q);
  int q = (int)(t % Q);
  if (e >= E) return;
  int s = src[e], d = dst[e];
  float w = dinv[s] * dinv[d];
  float4 hv = ((const float4*)(h + (size_t)s * D))[q];
  float* op = out + (size_t)d * D + q * 4;
  atomicAdd(op + 0, w * hv.x);
  atomicAdd(op + 1, w * hv.y);
  atomicAdd(op + 2, w * hv.z);
  atomicAdd(op + 3, w * hv.w);
}

// ---------------------------------------------------------------------------

static inline int cdiv(long long a, long long b) { return (int)((a + b - 1) / b); }

extern "C" void kernel_launch(void* const* d_in, const int* in_sizes, int n_in,
                              void* d_out, int out_size, void* d_ws,
                              size_t ws_size, hipStream_t stream) {
  const float* x   = (const float*)d_in[0];
  const int*   ei  = (const int*)d_in[1];   // [2, E] row-major
  const float* c   = (const float*)d_in[2];
  const float* W1  = (const float*)d_in[3];
  const float* b1  = (const float*)d_in[4];
  const float* W2  = (const float*)d_in[5];
  const float* b2  = (const float*)d_in[6];
  const float* Wmu = (const float*)d_in[7];
  const float* bmu = (const float*)d_in[8];
  const float* Wlv = (const float*)d_in[9];
  const float* blv = (const float*)d_in[10];
  float* outp = (float*)d_out;

  const int N = in_sizes[0] / 8;    // 100000
  const int E = in_sizes[1] / 2;    // 3200000
  const int Z = 32;
  const int ntiles = N / 16;        // 6250 (N % 16 == 0)
  const int* srcI = ei;
  const int* dstI = ei + E;

  // workspace layout (floats): dinv[N] | bufA[N*64] | bufB[N*64]
  float* dinv = (float*)d_ws;
  float* bufA = dinv + N;
  float* bufB = bufA + (size_t)N * 64;

  const int TB = 256;
  const int gemmBlocks = cdiv(ntiles, TB / 32);

  // 1) symmetric degree norm: deg = 1 (self loop) + scatter, then rsqrt
  fill_kernel<<<cdiv(N, TB), TB, 0, stream>>>(dinv, 1.0f, N);
  deg_scatter_kernel<<<cdiv(E, TB), TB, 0, stream>>>(dstI, dinv, E);
  rsqrt_kernel<<<cdiv(N, TB), TB, 0, stream>>>(dinv, N);

  // 2) layer 1: h = relu(Agg([x|c] @ W1) + b1)
  gemm1_kernel<<<gemmBlocks, TB, 0, stream>>>(x, c, W1, bufA, ntiles);
  agg_init_kernel<64><<<cdiv((long long)N * 16, TB), TB, 0, stream>>>(
      bufA, dinv, b1, bufB, N);
  scatter_kernel<64><<<cdiv((long long)E * 16, TB), TB, 0, stream>>>(
      srcI, dstI, dinv, bufA, bufB, E);
  relu_kernel<<<cdiv((long long)N * 64, TB), TB, 0, stream>>>(bufB,
                                                              N * 64);

  // 3) layer 2: h = relu(Agg(h @ W2) + b2)
  gemm_k64_kernel<4><<<gemmBlocks, TB, 0, stream>>>(bufB, W2, bufA, ntiles);
  agg_init_kernel<64><<<cdiv((long long)N * 16, TB), TB, 0, stream>>>(
      bufA, dinv, b2, bufB, N);  // overwrites bufB after gemm read (ordered)
  // NOTE: scatter must read the *pre-aggregation* h (bufA)
  scatter_kernel<64><<<cdiv((long long)E * 16, TB), TB, 0, stream>>>(
      srcI, dstI, dinv, bufA, bufB, E);
  relu_kernel<<<cdiv((long long)N * 64, TB), TB, 0, stream>>>(bufB, N * 64);

  // 4) mu / logvar heads: Agg(h @ W) + b  (no relu), straight into d_out
  float* muL = bufA;                  // [N x 32]
  float* lvL = bufA + (size_t)N * Z;  // [N x 32]
  gemm_k64_kernel<2><<<gemmBlocks, TB, 0, stream>>>(bufB, Wmu, muL, ntiles);
  gemm_k64_kernel<2><<<gemmBlocks, TB, 0, stream>>>(bufB, Wlv, lvL, ntiles);

  float* muOut = outp;
  float* lvOut = outp + (size_t)N * Z;
  agg_init_kernel<32><<<cdiv((long long)N * 8, TB), TB, 0, stream>>>(
      muL, dinv, bmu, muOut, N);
  scatter_kernel<32><<<cdiv((long long)E * 8, TB), TB, 0, stream>>>(
      srcI, dstI, dinv, muL, muOut, E);
  agg_init_kernel<32><<<cdiv((long long)N * 8, TB), TB, 0, stream>>>(
      lvL, dinv, blv, lvOut, N);
  scatter_kernel<32><<<cdiv((long long)E * 8, TB), TB, 0, stream>>>(
      srcI, dstI, dinv, lvL, lvOut, E);
}